// CapsConvLayer_4372276707736
// MI455X (gfx1250) — compile-verified
//
#include <hip/hip_runtime.h>
#include <hip/hip_bf16.h>

// CDNA5 (gfx1250) fused capsule-routing implementation, round 2.
// - u_hat [256,1152,10,16] (189 MB) is never materialized; both routing
//   contractions are GEMMs over x (9.4 MB) and W (5.9 MB), L2-resident.
// - fp32 WMMA (V_WMMA_F32_16X16X4_F32), one 16x16 tile per wave.
// - GEMM1 K-split 16x (160 -> 2560 waves) with deterministic partial-sum
//   reduction; dual accumulators break the serial WMMA RAW chain.

typedef __attribute__((ext_vector_type(2))) float v2f;
typedef __attribute__((ext_vector_type(8))) float v8f;

#define B_   256
#define K_   8
#define I_   1152
#define C_   10
#define O_   16
#define KK_  (K_ * I_)        // 9216  contraction dim for GEMM1 / rows of T
#define N_   (C_ * O_)        // 160   (class, out-capsule) fused dim
#define KQ1_ (KK_ / 4)        // 2304  total k-steps GEMM1
#define KSPLIT_ 16            // K-dim split for GEMM1
#define KQC_ (KQ1_ / KSPLIT_) // 144   k-steps per wave (even -> dual acc ok)
#define KQ2_ (B_ / 4)         // 64    k-steps GEMM2

// ---------------------------------------------------------------- utilities

__global__ void fill_zero(float* p, int n) {
    int i = blockIdx.x * blockDim.x + threadIdx.x;
    if (i < n) p[i] = 0.0f;
}

// xT[kk*256 + b] = x[b*9216 + kk]   (coalesced writes; reads hit L2)
__global__ void transpose_x(const float* __restrict__ x, float* __restrict__ xT) {
    int idx = blockIdx.x * blockDim.x + threadIdx.x;     // over KK_*B_
    if (idx >= KK_ * B_) return;
    int kk = idx / B_;
    int b  = idx % B_;
    xT[idx] = x[b * KK_ + kk];
}

// softmax over the I axis (axis 0 of b[I,C]); one block per class.
__global__ void softmax_I(const float* __restrict__ blog, float* __restrict__ cmat) {
    __shared__ float red[256];
    int cls = blockIdx.x;                 // 0..9
    int t = threadIdx.x;                  // 256 threads
    float m = -1e30f;
    for (int i = t; i < I_; i += 256) m = fmaxf(m, blog[i * C_ + cls]);
    red[t] = m; __syncthreads();
    for (int s = 128; s > 0; s >>= 1) { if (t < s) red[t] = fmaxf(red[t], red[t + s]); __syncthreads(); }
    float mx = red[0]; __syncthreads();
    float sum = 0.0f;
    for (int i = t; i < I_; i += 256) sum += expf(blog[i * C_ + cls] - mx);
    red[t] = sum; __syncthreads();
    for (int s = 128; s > 0; s >>= 1) { if (t < s) red[t] += red[t + s]; __syncthreads(); }
    float inv = 1.0f / red[0];
    for (int i = t; i < I_; i += 256)
        cmat[i * C_ + cls] = expf(blog[i * C_ + cls] - mx) * inv;
}

// Bp packed as [kq][n][4] with kk = kq*4+kr, so a lane's two B values are a
// contiguous float2:  Bp[(kq*160+n)*4 + kr] = c[i,cls] * W[i,cls,o,k]
__global__ void build_Bp(const float* __restrict__ W, const float* __restrict__ cmat,
                         float* __restrict__ Bp) {
    int idx = blockIdx.x * blockDim.x + threadIdx.x;     // over KK_*N_
    if (idx >= KK_ * N_) return;
    int n  = idx % N_;
    int kk = idx / N_;
    int k = kk / I_, i = kk % I_;
    int c = n >> 4,  o = n & 15;
    float w = W[i * (C_ * O_ * K_) + c * (O_ * K_) + o * K_ + k];
    Bp[(((kk >> 2) * N_) + n) * 4 + (kk & 3)] = w * cmat[i * C_ + c];
}

// --------------------------------------------------------------- GEMM1 (s)
// s[b, n] = sum_kk x[b, kk] * Bp[kk, n]   M=256 N=160 K=9216
// 160 tiles x KSPLIT_ waves = 2560 waves; each wave covers 144 k-steps with
// two independent accumulators, writes a deterministic partial tile.
__global__ void gemm_s_kernel(const float* __restrict__ x,
                              const float* __restrict__ Bp,
                              float* __restrict__ part) {   // [KSPLIT_][256][160]
    int wave = (blockIdx.x * blockDim.x + threadIdx.x) >> 5;   // 0..2559
    int lane = threadIdx.x & 31;
    int split = wave / 160;                // 0..15
    int tile  = wave % 160;
    int tileN = tile % (N_ / 16);          // 0..9
    int tileM = tile / (N_ / 16);          // 0..15
    int mrow = tileM * 16 + (lane & 15);
    int ncol = tileN * 16 + (lane & 15);
    int kh   = (lane >> 4) * 2;            // lanes 0-15: K=0,1 ; 16-31: K=2,3
    const float* aPtr = x  + (size_t)mrow * KK_ + kh;
    const float* bPtr = Bp + (size_t)ncol * 4 + kh;
    int kq0 = split * KQC_;
    v8f acc0 = {}, acc1 = {};
    for (int t = 0; t < KQC_; t += 2) {
        int kq = kq0 + t;
        v2f a0 = *(const v2f*)(aPtr + (size_t)kq * 4);
        v2f b0 = *(const v2f*)(bPtr + (size_t)kq * (N_ * 4));
        v2f a1 = *(const v2f*)(aPtr + (size_t)(kq + 1) * 4);
        v2f b1 = *(const v2f*)(bPtr + (size_t)(kq + 1) * (N_ * 4));
        acc0 = __builtin_amdgcn_wmma_f32_16x16x4_f32(
            false, a0, false, b0, (short)0, acc0, false, false);
        acc1 = __builtin_amdgcn_wmma_f32_16x16x4_f32(
            false, a1, false, b1, (short)0, acc1, false, false);
    }
    int half = lane >> 4, n = lane & 15;
    float* dst = part + (size_t)split * (B_ * N_);
#pragma unroll
    for (int r = 0; r < 8; ++r) {
        int m = tileM * 16 + r + half * 8;
        dst[m * N_ + tileN * 16 + n] = acc0[r] + acc1[r];
    }
}

// Deterministic fixed-order sum of the K-split partials -> s[256,160]
__global__ void reduce_partials(const float* __restrict__ part, float* __restrict__ s) {
    int idx = blockIdx.x * blockDim.x + threadIdx.x;     // B_*N_ = 40960
    if (idx >= B_ * N_) return;
    float a = 0.0f;
#pragma unroll
    for (int p = 0; p < KSPLIT_; ++p) a += part[(size_t)p * (B_ * N_) + idx];
    s[idx] = a;
}

// squash over the class dim; writes v directly into d_out ([256,10,16] flat).
__global__ void squash_kernel(const float* __restrict__ s, float* __restrict__ v) {
    int idx = blockIdx.x * blockDim.x + threadIdx.x;     // B_*O_ = 4096
    if (idx >= B_ * O_) return;
    int b = idx >> 4, o = idx & 15;
    const float* sp = s + b * N_ + o;
    float ns = 0.0f;
#pragma unroll
    for (int c = 0; c < C_; ++c) { float t = sp[c * O_]; ns += t * t; }
    float f = (ns / (1.0f + ns)) / (sqrtf(ns) + 1e-10f);
#pragma unroll
    for (int c = 0; c < C_; ++c) v[b * N_ + c * O_ + o] = sp[c * O_] * f;
}

// --------------------------------------------------------------- GEMM2 (T)
// T[kk, n] = sum_b xT[kk, b] * v[b, n]   M=9216 N=160 K=256 ; 5760 waves.
__global__ void gemm_T_kernel(const float* __restrict__ xT,
                              const float* __restrict__ v,
                              float* __restrict__ T) {
    int wave = (blockIdx.x * blockDim.x + threadIdx.x) >> 5;   // 0..5759
    int lane = threadIdx.x & 31;
    int tileN = wave % (N_ / 16);          // 0..9
    int tileM = wave / (N_ / 16);          // 0..575
    int mrow = tileM * 16 + (lane & 15);
    int ncol = tileN * 16 + (lane & 15);
    int kh   = (lane >> 4) * 2;
    const float* aPtr = xT + (size_t)mrow * B_ + kh;
    v8f acc0 = {}, acc1 = {};
    for (int bq = 0; bq < KQ2_; bq += 2) {
        int b0 = bq * 4 + kh;
        v2f a0 = *(const v2f*)(aPtr + (size_t)bq * 4);
        v2f a1 = *(const v2f*)(aPtr + (size_t)(bq + 1) * 4);
        v2f bb0, bb1;
        bb0.x = v[b0 * N_ + ncol];
        bb0.y = v[(b0 + 1) * N_ + ncol];
        bb1.x = v[(b0 + 4) * N_ + ncol];
        bb1.y = v[(b0 + 5) * N_ + ncol];
        acc0 = __builtin_amdgcn_wmma_f32_16x16x4_f32(
            false, a0, false, bb0, (short)0, acc0, false, false);
        acc1 = __builtin_amdgcn_wmma_f32_16x16x4_f32(
            false, a1, false, bb1, (short)0, acc1, false, false);
    }
    int half = lane >> 4, n = lane & 15;
#pragma unroll
    for (int r = 0; r < 8; ++r) {
        int m = tileM * 16 + r + half * 8;
        T[m * N_ + tileN * 16 + n] = acc0[r] + acc1[r];
    }
}

// b[i,c] += (1/B) * sum_{k,o} T[(k*I+i), c*16+o] * W[i,c,o,k]
__global__ void agr_kernel(const float* __restrict__ T, const float* __restrict__ W,
                           float* __restrict__ blog) {
    int idx = blockIdx.x * blockDim.x + threadIdx.x;     // I_*C_ = 11520
    if (idx >= I_ * C_) return;
    int i = idx / C_, c = idx % C_;
    float acc = 0.0f;
#pragma unroll
    for (int k = 0; k < K_; ++k) {
        const float* tp = T + (k * I_ + i) * N_ + c * O_;
        const float* wp = W + i * (C_ * O_ * K_) + c * (O_ * K_) + k;
#pragma unroll
        for (int o = 0; o < O_; ++o) acc += tp[o] * wp[o * K_];
    }
    blog[idx] += acc * (1.0f / (float)B_);
}

// ------------------------------------------------------------------- launch

extern "C" void kernel_launch(void* const* d_in, const int* in_sizes, int n_in,
                              void* d_out, int out_size, void* d_ws, size_t ws_size,
                              hipStream_t stream) {
    (void)in_sizes; (void)n_in; (void)out_size; (void)ws_size;
    const float* x = (const float*)d_in[0];     // [256, 8, 1152]
    const float* W = (const float*)d_in[1];     // [1152, 10, 16, 8]
    float* out = (float*)d_out;                 // [256, 10, 16, 1] flat = 40960

    float* ws   = (float*)d_ws;                 // ~24.1 MB total (L2-resident)
    float* xT   = ws;                           // 9216*256    = 2359296
    float* Bp   = xT   + (size_t)KK_ * B_;      // 2304*160*4  = 1474560
    float* sbuf = Bp   + (size_t)KK_ * N_;      // 256*160     = 40960
    float* T    = sbuf + (size_t)B_ * N_;       // 9216*160    = 1474560
    float* blog = T    + (size_t)KK_ * N_;      // 1152*10     = 11520
    float* cmat = blog + (size_t)I_ * C_;       // 1152*10     = 11520
    float* part = cmat + (size_t)I_ * C_;       // 16*256*160  = 655360

    transpose_x<<<(KK_ * B_ + 255) / 256, 256, 0, stream>>>(x, xT);
    fill_zero  <<<(I_ * C_ + 255) / 256, 256, 0, stream>>>(blog, I_ * C_);

    for (int it = 0; it < 3; ++it) {
        softmax_I<<<C_, 256, 0, stream>>>(blog, cmat);
        build_Bp <<<(KK_ * N_ + 255) / 256, 256, 0, stream>>>(W, cmat, Bp);
        // 2560 waves exactly: 640 blocks x 4 waves (EXEC all-ones for WMMA)
        gemm_s_kernel<<<640, 128, 0, stream>>>(x, Bp, part);
        reduce_partials<<<(B_ * N_ + 255) / 256, 256, 0, stream>>>(part, sbuf);
        squash_kernel<<<(B_ * O_ + 255) / 256, 256, 0, stream>>>(sbuf, out);
        if (it < 2) {   // last iteration's agreement update is discarded
            // 5760 waves exactly: 1440 blocks x 4 waves
            gemm_T_kernel<<<1440, 128, 0, stream>>>(xT, out, T);
            agr_kernel<<<(I_ * C_ + 255) / 256, 256, 0, stream>>>(T, W, blog);
        }
    }
}